// RoIAlignLayer_74165495267697
// MI455X (gfx1250) — compile-verified
//
#include <hip/hip_runtime.h>

// RoI-Align on MI455X (gfx1250), reformulated as a per-box dense f32 GEMM
// executed with V_WMMA_F32_16X16X4_F32.
//
// Boxes are <=224px with scale=1/16, so every ROI's bilinear corner taps fit
// in a 16x16 window of the 50x50 feature plane. Per box:
//     pooled[bin, c] = sum_p  W[bin, p] * Fwin[p, c],   p in 16x16 window
// W (49x256 positions, padded to 64 bins) is built in LDS by ds_add_f32
// scatter of the exact bilinear weights (mask & 1/cnt folded in); Fwin is a
// clamped window load. GEMM: M=64 x N=256 x K=256 in f32 WMMA (RNE).
//
// Round-3 changes:
//  * K-pair-packed LDS layouts: element (col, kk) lives at
//    [kk>>1][col*2 + (kk&1)], so each lane's WMMA fragment (K, K+1) is one
//    contiguous 8B-aligned v2f -> a single ds_load_b64 per operand per WMMA
//    (was 2 scalar ds_load_b32 + 2 v_add for the A side).
//  * Pair-row strides == 32 (mod 64): the two half-waves (pair rows r, r+1)
//    hit disjoint 32-bank sets -> conflict-free b64 fragment reads.
//  * sF double-buffered: chunk i+1's global->LDS window load overlaps chunk
//    i's WMMA loop.

typedef __attribute__((ext_vector_type(2))) float v2f;
typedef __attribute__((ext_vector_type(8))) float v8f;

#define CC   256      // channels
#define FH   50
#define FW   50
#define NT   512      // boxes
#define OHW  7
#define NBINS 49      // 7*7
#define WIN  16       // window edge
#define NPOS 256      // 16*16 window positions (= GEMM K)
#define NPAIR 128     // NPOS/2 K-pair rows
#define CHUNK 32      // channels per LDS chunk
#define NCHUNK (CC / CHUNK)
#define SF_STRIDE 96  // words per sF pair-row: 64 data + pad; 96 mod 64 == 32
#define SW_STRIDE 160 // words per sW pair-row: 128 data + pad; 160 mod 64 == 32

__global__ __launch_bounds__(256) void roi_align_wmma_kernel(
    const float* __restrict__ feat,   // (2,256,50,50)
    const float* __restrict__ boxes,  // (512,4)
    const int*   __restrict__ bidx,   // (512,)
    float* __restrict__ out)          // (512,256,7,7)
{
  extern __shared__ float smem[];
  float* sF0 = smem;                           // [NPAIR][SF_STRIDE]
  float* sF1 = smem + NPAIR * SF_STRIDE;       // [NPAIR][SF_STRIDE]
  float* sW  = smem + 2 * NPAIR * SF_STRIDE;   // [NPAIR][SW_STRIDE]

  const int box = blockIdx.x;
  const int tid = threadIdx.x;

  // ---- box geometry (uniform per block) ----
  const float sx = boxes[box*4+0] * 0.0625f;
  const float sy = boxes[box*4+1] * 0.0625f;
  const float ex = boxes[box*4+2] * 0.0625f;
  const float ey = boxes[box*4+3] * 0.0625f;
  const float roi_w = fmaxf(ex - sx, 1.0f);
  const float roi_h = fmaxf(ey - sy, 1.0f);
  const float bin_w = roi_w * (1.0f / OHW);
  const float bin_h = roi_h * (1.0f / OHW);
  const int gw = min(max((int)ceilf(bin_w), 1), 2);
  const int gh = min(max((int)ceilf(bin_h), 1), 2);
  const float inv_cnt = 1.0f / (float)(gw * gh);
  const int ox = (int)floorf(sx);
  const int oy = (int)floorf(sy);
  const int b  = bidx[box];

  // pair-packed index: (col, kk) -> [kk>>1][col*2 + (kk&1)]
  auto widx = [](int kk, int bin) { return (kk >> 1) * SW_STRIDE + bin * 2 + (kk & 1); };
  auto fidx = [](int kk, int ch)  { return (kk >> 1) * SF_STRIDE + ch * 2 + (kk & 1); };

  // ---- window loader: 16x16 window for CHUNK channels -> pair-packed LDS ----
  // Indices are clamped; positions past the image edge carry zero weight in W.
  auto load_chunk = [&](float* __restrict__ dst, int cb) {
    for (int task = tid; task < WIN * CHUNK; task += 256) {
      const int dy = task >> 5;          // 0..15
      const int cc = task & 31;          // 0..31
      const int yy = min(oy + dy, FH - 1);
      const float* src = feat + (((size_t)b * CC + (cb + cc)) * FH + yy) * FW;
      #pragma unroll
      for (int dx = 0; dx < WIN; ++dx) {
        const int xx = min(ox + dx, FW - 1);
        dst[fidx(dy * WIN + dx, cc)] = src[xx];
      }
    }
  };

  // ---- zero W ----
  for (int i = tid; i < NPAIR * SW_STRIDE; i += 256) sW[i] = 0.0f;
  __syncthreads();

  // ---- scatter exact bilinear weights into dense W (pair-packed in LDS) ----
  // tasks: 49 bins x 2x2 sample grid = 196
  if (tid < NBINS * 4) {
    const int bin = tid >> 2;
    const int iy = (tid >> 1) & 1, ix = tid & 1;
    if (iy < gh && ix < gw) {
      const int ph = bin / OHW, pw = bin % OHW;
      const float Y = sy + ph * bin_h + (iy + 0.5f) * bin_h / (float)gh;
      const float X = sx + pw * bin_w + (ix + 0.5f) * bin_w / (float)gw;
      if (Y > -1.0f && Y < (float)FH && X > -1.0f && X < (float)FW) {
        const float y = fmaxf(Y, 0.0f), x = fmaxf(X, 0.0f);
        const int y0 = min((int)floorf(y), FH - 1);
        const int x0 = min((int)floorf(x), FW - 1);
        const int y1 = min(y0 + 1, FH - 1), x1 = min(x0 + 1, FW - 1);
        const float ly = y - (float)y0, lx = x - (float)x0;
        const float hy = 1.0f - ly, hx = 1.0f - lx;
        const int p00 = (y0 - oy) * WIN + (x0 - ox);
        const int p01 = (y0 - oy) * WIN + (x1 - ox);
        const int p10 = (y1 - oy) * WIN + (x0 - ox);
        const int p11 = (y1 - oy) * WIN + (x1 - ox);
        atomicAdd(&sW[widx(p00, bin)], hy * hx * inv_cnt);
        atomicAdd(&sW[widx(p01, bin)], hy * lx * inv_cnt);
        atomicAdd(&sW[widx(p10, bin)], ly * hx * inv_cnt);
        atomicAdd(&sW[widx(p11, bin)], ly * lx * inv_cnt);
      }
    }
  }
  // prefill pipeline: chunk 0 into buffer 0 (overlaps weight scatter above)
  load_chunk(sF0, 0);

  // ---- per-wave tile assignment: 8 waves = 4 M-tiles x 2 N-tiles ----
  const int wv   = tid >> 5;
  const int lane = tid & 31;
  const int half = lane >> 4;     // lane group 0..15 vs 16..31
  const int l16  = lane & 15;
  const int mt   = wv & 3;        // bin tile (M)
  const int nt   = wv >> 2;       // channel tile (N) within chunk
  // lane-constant word offsets of this lane's fragment pair within a pair-row
  const int aOff = (mt * 16 + l16) * 2;
  const int bOff = (nt * 16 + l16) * 2;

  for (int ci = 0; ci < NCHUNK; ++ci) {
    __syncthreads();   // chunk ci resident; buffer (ci+1)&1 free for reuse

    // start loading chunk ci+1 into the other buffer (hidden under WMMA)
    if (ci + 1 < NCHUNK) load_chunk((ci & 1) ? sF0 : sF1, (ci + 1) * CHUNK);

    const float* __restrict__ sFc = (ci & 1) ? sF1 : sF0;

    // ---- D(16x16) = sum_k A(16x4) * B(4x16), K = 256 window positions ----
    // A f32 16x4 layout: VGPR0 = K0|K2 (per half-wave), VGPR1 = K1|K3, i.e.
    // lane needs the contiguous pair at pair-row (k>>1)+half -> one b64 load.
    v8f acc = {0.f, 0.f, 0.f, 0.f, 0.f, 0.f, 0.f, 0.f};
    #pragma unroll 8
    for (int k = 0; k < NPOS; k += 4) {
      const int pr = (k >> 1) + half;   // pair row for this half-wave
      const v2f a  = *(const v2f*)(sW  + pr * SW_STRIDE + aOff);
      const v2f bb = *(const v2f*)(sFc + pr * SF_STRIDE + bOff);
      acc = __builtin_amdgcn_wmma_f32_16x16x4_f32(
          /*neg_a=*/false, a, /*neg_b=*/false, bb,
          /*c_mod=*/(short)0, acc, /*reuse_a=*/false, /*reuse_b=*/false);
    }

    // ---- store: D VGPR r -> (M = r + 8*half, N = l16) ----
    const int c = ci * CHUNK + nt * 16 + l16;
    float* obase = out + ((size_t)box * CC + c) * NBINS;
    #pragma unroll
    for (int r = 0; r < 8; ++r) {
      const int bin = mt * 16 + r + half * 8;
      if (bin < NBINS) obase[bin] = acc[r];
    }
  }
}

extern "C" void kernel_launch(void* const* d_in, const int* in_sizes, int n_in,
                              void* d_out, int out_size, void* d_ws, size_t ws_size,
                              hipStream_t stream) {
  const float* feat  = (const float*)d_in[0];   // (2,256,50,50) f32
  const float* boxes = (const float*)d_in[1];   // (512,4) f32
  const int*   bidx  = (const int*)d_in[2];     // (512,) i32
  float* out = (float*)d_out;                   // (512,256,7,7) f32
  (void)in_sizes; (void)n_in; (void)out_size; (void)d_ws; (void)ws_size;

  const size_t lds_bytes =
      (size_t)(2 * NPAIR * SF_STRIDE + NPAIR * SW_STRIDE) * sizeof(float); // 176 KB
  roi_align_wmma_kernel<<<NT, 256, lds_bytes, stream>>>(feat, boxes, bidx, out);
}